// SimpleGreedy_2491081032184
// MI455X (gfx1250) — compile-verified
//
#include <hip/hip_runtime.h>
#include <stdint.h>

// Online bipartite matching (ranking algorithm), MI455X / gfx1250.
// One wave32 per batch row. Adjacency streamed HBM -> LDS with the CDNA5
// async copy path (global_load_async_to_lds_b128 + s_wait_asynccnt ring
// buffer, depth 8) so the serial per-step argmin never stalls on memory.
// Roofline: 134 MB adjacency stream @ 23.3 TB/s ~ 6 us; compute is trivial.

namespace {
constexpr int kB = 2048;   // batch
constexpr int kV = 128;    // online arrivals (sequential steps)
constexpr int kU = 128;    // offline nodes
constexpr int kWaves = 8;  // waves per block (256 threads)
constexpr int kDepth = 8;  // async ring depth (rows in flight per wave)
constexpr int kMatchedRank = 1 << 20;   // sentinel rank once matched
constexpr int kInfKey = 0x7FFFFFFF;     // not adjacent
}

__global__ __launch_bounds__(kWaves * 32)
void ranking_match_kernel(const float* __restrict__ weights,
                          const int* __restrict__ adj,
                          float* __restrict__ out)
{
  // Per-wave ring buffer of adjacency rows: 8 waves * 8 rows * 512 B = 32 KB.
  __shared__ __align__(16) int s_adj[kWaves][kDepth][kU];

  const int lane = threadIdx.x & 31;
  const int wave = threadIdx.x >> 5;
  const int b    = blockIdx.x * kWaves + wave;

  // ---- deterministic per-batch rank permutation of 1..129 -----------------
  // (reference uses JAX threefry Fisher-Yates; bit-exact reproduction is not
  //  possible off-line, so we use a structurally identical deterministic
  //  permutation: composition of two affine bijections mod 129 = 3*43.)
  uint32_t h = (uint32_t)b * 0x9E3779B9u;
  h ^= h >> 16; h *= 0x85EBCA6Bu; h ^= h >> 13; h *= 0xC2B2AE35u; h ^= h >> 16;
  uint32_t a1 = 1u + (h & 127u);
  while (a1 % 3u == 0u || a1 % 43u == 0u) a1++;
  uint32_t c1 = (h >> 7) % 129u;
  uint32_t a2 = 1u + ((h >> 16) & 127u);
  while (a2 % 3u == 0u || a2 % 43u == 0u) a2++;
  uint32_t c2 = (h >> 21) % 129u;

  // Lane l owns offline nodes u = 4l..4l+3 (option index o = u+1 in 1..128).
  int rnk[4];
  #pragma unroll
  for (int j = 0; j < 4; ++j) {
    uint32_t o = (uint32_t)(4 * lane + j + 1);
    uint32_t p = (a1 * o + c1) % 129u;
    p = (a2 * p + c2) % 129u;
    rnk[j] = (int)p + 1;                 // distinct values in 1..129
  }

  const int* adj_b = adj + (size_t)b * kV * kU;

  // ---- async prologue: stage rows 0..kDepth-1 into LDS --------------------
  #pragma unroll
  for (int r = 0; r < kDepth; ++r) {
    uint64_t ga = (uint64_t)(uintptr_t)(adj_b + r * kU + lane * 4);
    uint32_t la = (uint32_t)(uintptr_t)&s_adj[wave][r][lane * 4];
    asm volatile("global_load_async_to_lds_b128 %0, %1, off"
                 :: "v"(la), "v"(ga) : "memory");
  }

  float msize = 0.0f;
  int seqbuf[4];                          // 128 selections / 32 lanes
  const int skipKey = (2 * kU) << 9;      // option 0, rank 2*U (always ok)

  for (int v = 0; v < kV; ++v) {
    const int slot = v & (kDepth - 1);

    // Row v is the oldest outstanding async op. Steady state: allow the
    // kDepth-1 younger rows to stay in flight; tail: drain fully.
    if (v <= kV - kDepth) {
      asm volatile("s_wait_asynccnt 7" ::: "memory");
    } else {
      asm volatile("s_wait_asynccnt 0" ::: "memory");
    }

    const int4 a = *reinterpret_cast<const int4*>(&s_adj[wave][slot][lane * 4]);

    // key = (rank << 9) | option; invalid candidates -> +inf.
    int k0 = (a.x > 0) ? ((rnk[0] << 9) | (4 * lane + 1)) : kInfKey;
    int k1 = (a.y > 0) ? ((rnk[1] << 9) | (4 * lane + 2)) : kInfKey;
    int k2 = (a.z > 0) ? ((rnk[2] << 9) | (4 * lane + 3)) : kInfKey;
    int k3 = (a.w > 0) ? ((rnk[3] << 9) | (4 * lane + 4)) : kInfKey;
    int key = min(min(min(k0, k1), min(k2, k3)), skipKey);

    // The ds_load above must have retired before this slot is overwritten.
    asm volatile("s_wait_dscnt 0" ::: "memory");
    if (v + kDepth < kV) {
      uint64_t ga = (uint64_t)(uintptr_t)(adj_b + (v + kDepth) * kU + lane * 4);
      uint32_t la = (uint32_t)(uintptr_t)&s_adj[wave][slot][lane * 4];
      asm volatile("global_load_async_to_lds_b128 %0, %1, off"
                   :: "v"(la), "v"(ga) : "memory");
    }

    // wave32 min-reduction: 5 xor-shuffle steps.
    #pragma unroll
    for (int off = 16; off > 0; off >>= 1)
      key = min(key, __shfl_xor(key, off, 32));

    const int selected = key & 511;                 // 0 = skip, 1..128 = u+1
    if ((v & 31) == lane) seqbuf[v >> 5] = selected;

    if (selected > 0) {
      const int u = selected - 1;
      if ((u >> 2) == lane) {                       // owning lane
        rnk[u & 3] = kMatchedRank;                  // mark matched
        msize += weights[((size_t)b * kV + v) * kU + u];
      }
    }
  }

  // Sum partial sizes across the wave.
  #pragma unroll
  for (int off = 16; off > 0; off >>= 1)
    msize += __shfl_xor(msize, off, 32);
  if (lane == 0) out[b] = -msize;

  // Sequence output (coalesced: lane l wrote iterations v = 32*s + l).
  float* seqout = out + kB;
  #pragma unroll
  for (int s = 0; s < 4; ++s)
    seqout[(size_t)b * kV + s * 32 + lane] = (float)seqbuf[s];
}

extern "C" void kernel_launch(void* const* d_in, const int* in_sizes, int n_in,
                              void* d_out, int out_size, void* d_ws, size_t ws_size,
                              hipStream_t stream) {
  (void)in_sizes; (void)n_in; (void)d_ws; (void)ws_size; (void)out_size;
  const float* weights = (const float*)d_in[0];
  const int*   adjacency = (const int*)d_in[1];
  float* out = (float*)d_out;

  dim3 grid(kB / kWaves);      // 256 blocks
  dim3 block(kWaves * 32);     // 256 threads = 8 wave32
  hipLaunchKernelGGL(ranking_match_kernel, grid, block, 0, stream,
                     weights, adjacency, out);
}